// ssd_56848187129935
// MI455X (gfx1250) — compile-verified
//
#include <hip/hip_runtime.h>
#include <cstdint>
#include <math.h>

// ---------------- problem constants (match reference) ----------------
#define B_IMG   64
#define NPRIOR  8732
#define NCLS    21
#define NFG     20      // foreground classes
#define TOPK    200
#define CONF_TH 0.01f
#define NMS_TH  0.45f
#define VAR0    0.1f
#define VAR1    0.2f
#define NTHR    256     // 8 wave32 per block

#define NROWS   (B_IMG * NPRIOR)     // 558848 softmax rows
#define NTILES  (NROWS / 16)         // 34928 (exact)
#define SBLOCKS (NTILES / 8)         // 4366 blocks, one 16-row tile per wave

typedef __attribute__((ext_vector_type(2))) float v2f;
typedef __attribute__((ext_vector_type(8))) float v8f;

// ---------------- CDNA5 async global->LDS helpers ----------------
__device__ __forceinline__ void async_g2lds_b128(unsigned lds_byte_off,
                                                 const void* gaddr) {
  asm volatile("global_load_async_to_lds_b128 %0, %1, off"
               :: "v"(lds_byte_off), "v"(gaddr)
               : "memory");
}
__device__ __forceinline__ void wait_asynccnt0() {
  asm volatile("s_wait_asynccnt 0" ::: "memory");
}
template <typename T>
__device__ __forceinline__ unsigned lds_off(const T* p) {
  return (unsigned)(uintptr_t)p;   // low 32 bits of flat addr = LDS offset
}

// ---------------- kernel 1: clear the output every launch ----------------
__global__ void ssd_zero_kernel(float* __restrict__ out, int n) {
  int i = blockIdx.x * blockDim.x + threadIdx.x;
  if (i < n) out[i] = 0.0f;
}

// ---------------- kernel 2: softmax stats (max, denom) via WMMA ----------
// One wave32 per 16-row tile. Lane L and lane L+16 cooperate on row L:
// A-matrix 16x4 f32 layout is lanes 0-15 -> K=0,1 ; lanes 16-31 -> K=2,3.
// denom(row) = sum_k exp(x_k - max) is computed as 6 chained
// V_WMMA_F32_16X16X4_F32 with B = ones (rank-deficient on purpose: the
// matrix pipe does the K-reduction and co-executes with the VALU exp's).
__global__ __launch_bounds__(NTHR) void ssd_softmax_stats_kernel(
    const float* __restrict__ conf,   // (B*P, 21)
    float* __restrict__ wmax,         // (B*P)
    float* __restrict__ wden)         // (B*P)
{
  __shared__ float s_x[8 * 256];      // 8 waves * (8 VGPRs * 32 lanes)

  const int tid  = threadIdx.x;
  const int lane = tid & 31;
  const int wid  = tid >> 5;
  const int half = lane >> 4;         // 0: classes k%4 in {0,1}; 1: {2,3}
  const int r    = lane & 15;         // row within tile
  const int tile = blockIdx.x * 8 + wid;
  const long row = (long)tile * 16 + r;

  // Load this row's 21 logits (lane pair loads redundantly; L2-resident).
  const float* v = conf + row * NCLS;
  float x[NCLS];
  #pragma unroll
  for (int k = 0; k < NCLS; ++k) x[k] = v[k];
  float m = x[0];
  #pragma unroll
  for (int k = 1; k < NCLS; ++k) m = fmaxf(m, x[k]);

  // 6 chained WMMAs: chunk t covers classes 4t .. 4t+3 (pad >=21 with 0).
  v2f bones; bones.x = 1.0f; bones.y = 1.0f;
  v8f c = {};
  #pragma unroll
  for (int t = 0; t < 6; ++t) {
    const int k0 = 4 * t + 2 * half;
    v2f a;
    a.x = (k0     < NCLS) ? expf(x[k0]     - m) : 0.0f;
    a.y = (k0 + 1 < NCLS) ? expf(x[k0 + 1] - m) : 0.0f;
    c = __builtin_amdgcn_wmma_f32_16x16x4_f32(
        /*neg_a=*/false, a, /*neg_b=*/false, bones,
        /*c_mod=*/(short)0, c, /*reuse_a=*/false, /*reuse_b=*/false);
  }

  // Extract row sums from the C tile: D[M][N] sits at VGPR (M&7),
  // lanes 0-15 for M<8 and lanes 16-31 for M>=8. Bounce through LDS.
  #pragma unroll
  for (int q = 0; q < 8; ++q) s_x[wid * 256 + q * 32 + lane] = c[q];
  __syncthreads();   // fence the cross-lane LDS exchange

  if (half == 0) {   // lanes 0-15 publish rows 0-15 of this tile
    const float den = s_x[wid * 256 + (r & 7) * 32 + ((r >= 8) ? 16 : 0)];
    wmax[row] = m;
    wden[row] = den;
  }
}

// ---------------- kernel 3: one block per (image, fg-class) --------------
__global__ __launch_bounds__(NTHR) void ssd_detect_kernel(
    const float* __restrict__ loc,    // (B, P, 4)
    const float* __restrict__ conf,   // (B, P, 21)
    const float* __restrict__ prior,  // (P, 4)
    const float* __restrict__ wmax,   // (B*P)
    const float* __restrict__ wden,   // (B*P)
    float* __restrict__ out)          // (B, 21, TOPK, 5)
{
  __shared__ float s_scores[NPRIOR];          // 34928 B
  __shared__ float s_wv[8];
  __shared__ int   s_wi[8];
  __shared__ float s_topv[TOPK];
  __shared__ int   s_topi[TOPK];
  __shared__ __align__(16) float s_gather[TOPK * 8]; // async dst: loc[4]+prior[4]
  __shared__ float s_box[TOPK][4];
  __shared__ int   s_keep[TOPK];
  __shared__ int   s_pos[TOPK];

  const int tid  = threadIdx.x;
  const int lane = tid & 31;
  const int wid  = tid >> 5;
  const int bc   = blockIdx.x;
  const int b    = bc / NFG;
  const int c    = bc % NFG;           // conf class index is c+1
  const float NEG_INF = -__builtin_inff();

  // ---- 1) scores from precomputed softmax stats + threshold ----
  const size_t rowbase = (size_t)b * NPRIOR;
  for (int p = tid; p < NPRIOR; p += NTHR) {
    const float mm = wmax[rowbase + p];
    const float dd = wden[rowbase + p];
    const float xx = conf[(rowbase + p) * NCLS + (c + 1)];
    const float sc = expf(xx - mm) / dd;
    s_scores[p] = (sc > CONF_TH) ? sc : NEG_INF;
  }
  __syncthreads();

  // ---- 2) top-200: register-cached segment maxima + wave32 butterfly ----
  // Thread t owns priors p = t (mod 256); removing the winner invalidates
  // only the owner's cached maximum, so only one thread rescans per round.
  float lv = NEG_INF;
  int   li = 0x7fffffff;
  for (int p = tid; p < NPRIOR; p += NTHR) {
    const float vv = s_scores[p];
    if (vv > lv) { lv = vv; li = p; }   // first-seen => lowest index on ties
  }
  for (int it = 0; it < TOPK; ++it) {
    float v = lv; int i = li;
    #pragma unroll
    for (int mlt = 16; mlt >= 1; mlt >>= 1) {
      const float ov = __shfl_xor(v, mlt, 32);
      const int   oi = __shfl_xor(i, mlt, 32);
      if (ov > v || (ov == v && oi < i)) { v = ov; i = oi; }
    }
    if (lane == 0) { s_wv[wid] = v; s_wi[wid] = i; }
    __syncthreads();
    float wv = s_wv[0]; int wi = s_wi[0];
    #pragma unroll
    for (int w = 1; w < 8; ++w) {
      const float ov = s_wv[w]; const int oi = s_wi[w];
      if (ov > wv || (ov == wv && oi < wi)) { wv = ov; wi = oi; }
    }
    if (tid == 0) {
      s_topv[it] = wv;
      s_topi[it] = (wi == 0x7fffffff) ? 0 : wi;
    }
    if (wi != 0x7fffffff && (wi & (NTHR - 1)) == tid) {
      s_scores[wi] = NEG_INF;           // remove winner, refresh owner cache
      lv = NEG_INF; li = 0x7fffffff;
      for (int p = tid; p < NPRIOR; p += NTHR) {
        const float xv = s_scores[p];
        if (xv > lv) { lv = xv; li = p; }
      }
    }
    __syncthreads();
  }

  // ---- 3) async global->LDS gather of the 200 candidate loc/prior rows ----
  if (tid < TOPK) {
    const int idx = s_topi[tid];
    const float* lp = loc   + ((size_t)b * NPRIOR + idx) * 4;
    const float* pp = prior + (size_t)idx * 4;
    const unsigned base = lds_off(&s_gather[0]) + (unsigned)tid * 32u;
    async_g2lds_b128(base,       lp);   // 16B loc row   -> LDS
    async_g2lds_b128(base + 16u, pp);   // 16B prior row -> LDS
  }
  wait_asynccnt0();
  __syncthreads();

  // ---- 4) decode, keep = valid ----
  if (tid < TOPK) {
    const float lx = s_gather[tid * 8 + 0], ly = s_gather[tid * 8 + 1];
    const float lw = s_gather[tid * 8 + 2], lh = s_gather[tid * 8 + 3];
    const float px = s_gather[tid * 8 + 4], py = s_gather[tid * 8 + 5];
    const float pw = s_gather[tid * 8 + 6], ph = s_gather[tid * 8 + 7];
    const float cx = px + lx * VAR0 * pw;
    const float cy = py + ly * VAR0 * ph;
    const float w  = pw * expf(lw * VAR1);
    const float h  = ph * expf(lh * VAR1);
    const float x1 = cx - w * 0.5f;
    const float y1 = cy - h * 0.5f;
    s_box[tid][0] = x1;
    s_box[tid][1] = y1;
    s_box[tid][2] = x1 + w;
    s_box[tid][3] = y1 + h;
    s_keep[tid] = (s_topv[tid] > CONF_TH) ? 1 : 0;
  }
  __syncthreads();

  // ---- 5) greedy NMS: serial pivot i, parallel over j > i ----
  for (int i = 0; i < TOPK; ++i) {
    const int active = s_keep[i];       // keep[i] implies valid[i]
    if (active && tid > i && tid < TOPK && s_keep[tid]) {
      const float xi1 = fmaxf(s_box[i][0], s_box[tid][0]);
      const float yi1 = fmaxf(s_box[i][1], s_box[tid][1]);
      const float xi2 = fminf(s_box[i][2], s_box[tid][2]);
      const float yi2 = fminf(s_box[i][3], s_box[tid][3]);
      const float inter = fmaxf(xi2 - xi1, 0.0f) * fmaxf(yi2 - yi1, 0.0f);
      const float ai = (s_box[i][2] - s_box[i][0]) * (s_box[i][3] - s_box[i][1]);
      const float aj = (s_box[tid][2] - s_box[tid][0]) * (s_box[tid][3] - s_box[tid][1]);
      const float iou = inter / (ai + aj - inter);  // NaN>th false, like jnp
      if (iou > NMS_TH) s_keep[tid] = 0;
    }
    __syncthreads();
  }

  // ---- 6) stable compaction (kept first, original order), then emit ----
  if (tid == 0) {
    int cnt = 0;
    for (int k = 0; k < TOPK; ++k) {
      s_pos[k] = s_keep[k] ? cnt : -1;
      cnt += s_keep[k];
    }
  }
  __syncthreads();

  float* base_out = out + (((size_t)b * NCLS) + (size_t)(c + 1)) * TOPK * 5;
  if (tid < TOPK && s_keep[tid]) {
    const int r = s_pos[tid];
    base_out[r * 5 + 0] = s_topv[tid];
    base_out[r * 5 + 1] = s_box[tid][0];
    base_out[r * 5 + 2] = s_box[tid][1];
    base_out[r * 5 + 3] = s_box[tid][2];
    base_out[r * 5 + 4] = s_box[tid][3];
  }
}

// ---------------- host entry ----------------
extern "C" void kernel_launch(void* const* d_in, const int* in_sizes, int n_in,
                              void* d_out, int out_size, void* d_ws, size_t ws_size,
                              hipStream_t stream) {
  (void)in_sizes; (void)n_in; (void)ws_size;
  const float* loc   = (const float*)d_in[0];  // (64, 8732, 4)
  const float* conf  = (const float*)d_in[1];  // (64, 8732, 21)
  const float* prior = (const float*)d_in[2];  // (8732, 4)
  float* out  = (float*)d_out;                 // (64, 21, 200, 5)
  float* wmax = (float*)d_ws;                  // (B*P) row max
  float* wden = wmax + NROWS;                  // (B*P) softmax denominator

  ssd_zero_kernel<<<(out_size + NTHR - 1) / NTHR, NTHR, 0, stream>>>(out, out_size);
  ssd_softmax_stats_kernel<<<SBLOCKS, NTHR, 0, stream>>>(conf, wmax, wden);
  ssd_detect_kernel<<<B_IMG * NFG, NTHR, 0, stream>>>(loc, conf, prior,
                                                      wmax, wden, out);
}